// DeformableInception_18502719111679
// MI455X (gfx1250) — compile-verified
//
#include <hip/hip_runtime.h>

// ---------------------------------------------------------------------------
// DeformableInception on MI455X (gfx1250), fp32 end-to-end via
// V_WMMA_F32_16X16X4_F32 (exact fp32 matrix pipe; K=2304..4608 reductions
// rule out bf16 inputs against an fp32 reference).
//
// v2 changes vs v1:
//  * Weights pre-transposed once into K-major scratch (coalesced A loads;
//    w_f padded M 324->336 so the 3x3 inner loop has no masking).
//  * N-tile widened to 32 pixels/WG: each A fragment feeds 2 WMMAs
//    (1 global load per WMMA instead of 2).
//  * B panels stored K-paired in LDS so each B fragment is one ds_load_b64.
//
// Stage plan (256 threads = 8 wave32 per WG, 1024 WGs per GEMM kernel):
//   T:  transpose w_dc0/w_dc1/w_cc/w_f into d_ws (one-time, ~10MB)
//   K1: deform_conv (x2)  : M=256,  K=2304, N=32768  -> o0/o1
//   K2: 1x1 conv on concat: M=512,  K=512,  N=32768  -> h
//   K3: 3x3 conv pad=1    : M=336p, K=4608, N=32768  -> d_out (store M<324)
// ---------------------------------------------------------------------------

typedef __attribute__((ext_vector_type(2))) float v2f;
typedef __attribute__((ext_vector_type(8))) float v8f;

#define WMMA_F32(a, b, c) \
  __builtin_amdgcn_wmma_f32_16x16x4_f32(false, (a), false, (b), (short)0, (c), false, false)

// ---------------------------------------------------------------------------
// One-time weight transpose: out[k*Mpad + m] = (m < M) ? in[m*K + k] : 0
// ---------------------------------------------------------------------------
__launch_bounds__(256)
__global__ void transpose_pad(const float* __restrict__ in, float* __restrict__ out,
                              int M, int K, int Mpad)
{
  const int idx = blockIdx.x * 256 + threadIdx.x;
  if (idx < K * Mpad) {
    const int k = idx / Mpad, m = idx - k * Mpad;
    out[idx] = (m < M) ? in[(size_t)m * K + k] : 0.0f;
  }
}

// ---------------------------------------------------------------------------
// Kernel 1: deformable conv, implicit GEMM. WG = 32 pixels x all 256 M.
//   wT layout: wT[(c*9+tap)*256 + m]
// ---------------------------------------------------------------------------
__launch_bounds__(256)
__global__ void deform_conv_wmma(const float* __restrict__ x,
                                 const float* __restrict__ off,
                                 const float* __restrict__ wT,
                                 float* __restrict__ out)
{
  constexpr int C = 256, HW = 4096, KK = 9;
  __shared__ int   s_idx[KK * 32 * 4];
  __shared__ float s_wgt[KK * 32 * 4];
  __shared__ __align__(16) float s_B[(C / 2) * 32 * 2];   // paired panel, 32 KB

  const int b    = blockIdx.x >> 7;            // 128 tiles of 32 px per image
  const int pix0 = (blockIdx.x & 127) * 32;
  const int t    = threadIdx.x;

  // --- Stage 1: bilinear corner indices & weights (9 taps x 32 px) ---------
  for (int p = t; p < KK * 32; p += 256) {
    const int tap = p >> 5, px = p & 31;
    const int pix = pix0 + px;
    const int hh = pix >> 6, ww = pix & 63;
    const int ky = tap / 3, kx = tap % 3;
    const float dy = off[((b * 18 + tap * 2    ) << 12) + pix];
    const float dx = off[((b * 18 + tap * 2 + 1) << 12) + pix];
    const float py  = (float)(hh + ky - 1) + dy;
    const float pxx = (float)(ww + kx - 1) + dx;
    const float y0f = floorf(py), x0f = floorf(pxx);
    const float fy = py - y0f, fx = pxx - x0f;
    const int y0 = (int)y0f, x0 = (int)x0f;
#pragma unroll
    for (int i = 0; i < 4; ++i) {
      const int yi = y0 + (i >> 1), xi = x0 + (i & 1);
      const bool ok = (yi >= 0) & (yi < 64) & (xi >= 0) & (xi < 64);
      const float wy = (i >> 1) ? fy : (1.0f - fy);
      const float wx = (i & 1) ? fx : (1.0f - fx);
      s_idx[p * 4 + i] = ok ? (yi * 64 + xi) : 0;       // clamp, weight masks
      s_wgt[p * 4 + i] = ok ? (wy * wx) : 0.0f;
    }
  }
  __syncthreads();

  const int lane = t & 31;
  const int wv   = t >> 5;
  const int m0   = wv * 32;                    // wave: two M=16 tiles
  const int n    = lane & 15;
  const int kb   = (lane < 16) ? 0 : 2;        // even -> (kb,kb+1) is a pair

  v8f acc[2][2] = {};                          // [mtile][ntile]
  const float* xb = x + (size_t)b * C * HW;

  for (int tap = 0; tap < KK; ++tap) {
    // --- Stage 2: gather-sample B panel: 256 ch x 32 px (paired layout) ---
    {
      const int px = t & 31, cb = t >> 5;      // cb in 0..7
      const int base = (tap * 32 + px) * 4;
      const int   i0 = s_idx[base + 0], i1 = s_idx[base + 1];
      const int   i2 = s_idx[base + 2], i3 = s_idx[base + 3];
      const float w0 = s_wgt[base + 0], w1 = s_wgt[base + 1];
      const float w2 = s_wgt[base + 2], w3 = s_wgt[base + 3];
#pragma unroll 4
      for (int j = 0; j < 32; ++j) {
        const int c = cb + (j << 3);
        const float* xc = xb + c * HW;
        s_B[(c >> 1) * 64 + (px << 1) + (c & 1)] =
            xc[i0] * w0 + xc[i1] * w1 + xc[i2] * w2 + xc[i3] * w3;
      }
    }
    __syncthreads();

    // --- Stage 3: K-loop, 64 chunks of 4 channels -------------------------
    for (int cblk = 0; cblk < 64; ++cblk) {
      const int c4 = cblk << 2;
      const int prow = ((c4 + kb) >> 1) * 64;  // paired-row base
      const v2f bf0 = *reinterpret_cast<const v2f*>(&s_B[prow + (n << 1)]);
      const v2f bf1 = *reinterpret_cast<const v2f*>(&s_B[prow + ((16 + n) << 1)]);
      const int kd = (c4 + kb) * KK + tap;     // K index = c*9 + tap
      v2f a0, a1;                              // coalesced: lanes n contiguous
      a0.x = wT[(size_t)kd * 256 + m0 + n];
      a0.y = wT[(size_t)(kd + KK) * 256 + m0 + n];
      a1.x = wT[(size_t)kd * 256 + m0 + 16 + n];
      a1.y = wT[(size_t)(kd + KK) * 256 + m0 + 16 + n];
      acc[0][0] = WMMA_F32(a0, bf0, acc[0][0]);
      acc[0][1] = WMMA_F32(a0, bf1, acc[0][1]);
      acc[1][0] = WMMA_F32(a1, bf0, acc[1][0]);
      acc[1][1] = WMMA_F32(a1, bf1, acc[1][1]);
    }
    __syncthreads();
  }

  // --- Stage 4: store D tiles ----------------------------------------------
#pragma unroll
  for (int mt = 0; mt < 2; ++mt)
#pragma unroll
    for (int r = 0; r < 8; ++r) {
      const int mm = m0 + mt * 16 + r + ((lane < 16) ? 0 : 8);
      out[(((size_t)b * 256 + mm) << 12) + pix0 + n]      = acc[mt][0][r];
      out[(((size_t)b * 256 + mm) << 12) + pix0 + 16 + n] = acc[mt][1][r];
    }
}

// ---------------------------------------------------------------------------
// Kernel 2: 1x1 conv over (virtual) concat: M=512, K=512.
//   wccT layout: wccT[k*512 + m]
// ---------------------------------------------------------------------------
__launch_bounds__(256)
__global__ void conv1x1_wmma(const float* __restrict__ o0,
                             const float* __restrict__ o1,
                             const float* __restrict__ wccT,
                             const float* __restrict__ bcc,
                             float* __restrict__ hout)
{
  __shared__ __align__(16) float s_B[256 * 64];   // 512 ch x 32 px paired, 64 KB

  const int b    = blockIdx.x >> 7;
  const int pix0 = (blockIdx.x & 127) * 32;
  const int t    = threadIdx.x;

  {
    const int px = t & 31, cb = t >> 5;
#pragma unroll 4
    for (int j = 0; j < 64; ++j) {               // c = cb + 8j, uniform split
      const int c = cb + (j << 3);
      const float* src = (c < 256) ? (o0 + (((size_t)b * 256 + c) << 12))
                                   : (o1 + (((size_t)b * 256 + (c - 256)) << 12));
      s_B[(c >> 1) * 64 + (px << 1) + (c & 1)] = src[pix0 + px];
    }
  }
  __syncthreads();

  const int lane = t & 31, wv = t >> 5;
  const int n = lane & 15, kb = (lane < 16) ? 0 : 2;

  v8f acc[4][2] = {};                            // M tiles wv+8j x 2 N tiles
  for (int cblk = 0; cblk < 128; ++cblk) {
    const int c4 = cblk << 2;
    const int prow = ((c4 + kb) >> 1) * 64;
    const v2f bf0 = *reinterpret_cast<const v2f*>(&s_B[prow + (n << 1)]);
    const v2f bf1 = *reinterpret_cast<const v2f*>(&s_B[prow + ((16 + n) << 1)]);
#pragma unroll
    for (int j = 0; j < 4; ++j) {
      const int row = (wv + 8 * j) * 16 + n;
      v2f a;
      a.x = wccT[(size_t)(c4 + kb) * 512 + row];
      a.y = wccT[(size_t)(c4 + kb + 1) * 512 + row];
      acc[j][0] = WMMA_F32(a, bf0, acc[j][0]);
      acc[j][1] = WMMA_F32(a, bf1, acc[j][1]);
    }
  }

#pragma unroll
  for (int j = 0; j < 4; ++j)
#pragma unroll
    for (int r = 0; r < 8; ++r) {
      const int mm = (wv + 8 * j) * 16 + r + ((lane < 16) ? 0 : 8);
      const float bias = bcc[mm];
      hout[(((size_t)b * 512 + mm) << 12) + pix0 + n]      = acc[j][0][r] + bias;
      hout[(((size_t)b * 512 + mm) << 12) + pix0 + 16 + n] = acc[j][1][r] + bias;
    }
}

// ---------------------------------------------------------------------------
// Kernel 3: 3x3 conv pad=1: M=336 (padded, zero rows), K=512*9.
//   wfT layout: wfT[(c*9+tap)*336 + m], rows m>=324 are zero.
// ---------------------------------------------------------------------------
__launch_bounds__(256)
__global__ void conv3x3_wmma(const float* __restrict__ hin,
                             const float* __restrict__ wfT,
                             const float* __restrict__ bf_,
                             float* __restrict__ out)
{
  constexpr int NCLS = 324, MP = 336, KK = 9;
  __shared__ __align__(16) float s_B[256 * 64];  // 512 ch x 32 px paired, 64 KB

  const int b    = blockIdx.x >> 7;
  const int pix0 = (blockIdx.x & 127) * 32;
  const int hrow = pix0 >> 6, wcol0 = pix0 & 63;
  const int t    = threadIdx.x;
  const int lane = t & 31, wv = t >> 5;
  const int n = lane & 15, kb = (lane < 16) ? 0 : 2;

  v8f acc[3][2] = {};                            // M tiles wv+8j (<21) x 2 N

  for (int tap = 0; tap < KK; ++tap) {
    const int ky = tap / 3, kx = tap % 3;
    {
      const int px = t & 31, cb = t >> 5;
      const int yy = hrow + ky - 1;
      const int xx = wcol0 + px + kx - 1;
      const bool ok = (yy >= 0) & (yy < 64) & (xx >= 0) & (xx < 64);
      const int   sidx = ok ? (yy * 64 + xx) : 0;
      const float msk  = ok ? 1.0f : 0.0f;
#pragma unroll 4
      for (int j = 0; j < 64; ++j) {
        const int c = cb + (j << 3);
        s_B[(c >> 1) * 64 + (px << 1) + (c & 1)] =
            hin[(((size_t)b * 512 + c) << 12) + sidx] * msk;
      }
    }
    __syncthreads();

    for (int cblk = 0; cblk < 128; ++cblk) {
      const int c4 = cblk << 2;
      const int prow = ((c4 + kb) >> 1) * 64;
      const v2f bf0 = *reinterpret_cast<const v2f*>(&s_B[prow + (n << 1)]);
      const v2f bf1 = *reinterpret_cast<const v2f*>(&s_B[prow + ((16 + n) << 1)]);
      const int kd = (c4 + kb) * KK + tap;
#pragma unroll
      for (int j = 0; j < 3; ++j) {
        const int mt = wv + 8 * j;               // wave-uniform branch
        if (mt < 21) {
          v2f a;
          a.x = wfT[(size_t)kd * MP + mt * 16 + n];
          a.y = wfT[(size_t)(kd + KK) * MP + mt * 16 + n];
          acc[j][0] = WMMA_F32(a, bf0, acc[j][0]);
          acc[j][1] = WMMA_F32(a, bf1, acc[j][1]);
        }
      }
    }
    __syncthreads();
  }

#pragma unroll
  for (int j = 0; j < 3; ++j) {
    const int mt = wv + 8 * j;
    if (mt < 21) {
#pragma unroll
      for (int r = 0; r < 8; ++r) {
        const int mm = mt * 16 + r + ((lane < 16) ? 0 : 8);
        if (mm < NCLS) {
          const float bias = bf_[mm];
          out[(((size_t)b * NCLS + mm) << 12) + pix0 + n]      = acc[j][0][r] + bias;
          out[(((size_t)b * NCLS + mm) << 12) + pix0 + 16 + n] = acc[j][1][r] + bias;
        }
      }
    }
  }
}

// ---------------------------------------------------------------------------
extern "C" void kernel_launch(void* const* d_in, const int* in_sizes, int n_in,
                              void* d_out, int out_size, void* d_ws, size_t ws_size,
                              hipStream_t stream)
{
  const float* x    = (const float*)d_in[0];
  const float* dm0  = (const float*)d_in[1];
  const float* dm1  = (const float*)d_in[2];
  const float* wdc0 = (const float*)d_in[3];
  const float* wdc1 = (const float*)d_in[4];
  const float* wcc  = (const float*)d_in[5];
  const float* bcc  = (const float*)d_in[6];
  const float* wf   = (const float*)d_in[7];
  const float* bff  = (const float*)d_in[8];
  float* out = (float*)d_out;

  // Scratch layout (floats): o0 | o1 | h | wT0 | wT1 | wccT | wfT  (~146 MB)
  float* o0   = (float*)d_ws;
  float* o1   = o0   + (size_t)8 * 256 * 4096;
  float* hb   = o1   + (size_t)8 * 256 * 4096;
  float* wT0  = hb   + (size_t)8 * 512 * 4096;
  float* wT1  = wT0  + (size_t)2304 * 256;
  float* wccT = wT1  + (size_t)2304 * 256;
  float* wfT  = wccT + (size_t)512 * 512;

  const dim3 blk(256);

  // One-time weight transposes (K-major, coalesced A loads downstream)
  transpose_pad<<<dim3((2304 * 256 + 255) / 256), blk, 0, stream>>>(wdc0, wT0, 256, 2304, 256);
  transpose_pad<<<dim3((2304 * 256 + 255) / 256), blk, 0, stream>>>(wdc1, wT1, 256, 2304, 256);
  transpose_pad<<<dim3((512 * 512 + 255) / 256), blk, 0, stream>>>(wcc, wccT, 512, 512, 512);
  transpose_pad<<<dim3((4608 * 336 + 255) / 256), blk, 0, stream>>>(wf, wfT, 324, 4608, 336);

  const dim3 grid(8 * 128);   // batch x 32-pixel tiles

  deform_conv_wmma<<<grid, blk, 0, stream>>>(x, dm0, wT0, o0);
  deform_conv_wmma<<<grid, blk, 0, stream>>>(x, dm1, wT1, o1);
  conv1x1_wmma<<<grid, blk, 0, stream>>>(o0, o1, wccT, bcc, hb);
  conv3x3_wmma<<<grid, blk, 0, stream>>>(hb, wfT, bff, out);
}